// PointPillarScatter_5669356834440
// MI455X (gfx1250) — compile-verified
//
#include <hip/hip_runtime.h>
#include <hip/hip_bf16.h>
#include <stdint.h>

// ---------------------------------------------------------------------------
// PointPillarScatter for MI455X (gfx1250)
//   B=8, C=64, NX=512, NY=512, NZ=1, NPB=16000
//   out[b][c][g] = pf[p][c] where g = z + y*NX + x from coords row p, else 0
// Traffic: 512MB dense zero write (dominant) + 32MB coalesced read + 32MB
// scattered 4B writes.  ~23us HBM floor at 23.3 TB/s.
// ---------------------------------------------------------------------------

#define PP_B   8
#define PP_C   64
#define PP_NX  512
#define PP_NY  512
#define PP_G   (PP_NX * PP_NY)   // 262144 cells per batch plane

// ---------------------------------------------------------------------------
// Kernel 1: zero-fill d_out using CDNA5 async LDS->global streaming stores.
// Each lane owns a 16B zero block in LDS and issues
//   global_store_async_from_lds_b128 v[addr64], v_ldsoff, off
// per 16B chunk of the output (ASYNCcnt-tracked; HW throttles at 63
// outstanding).  Drained with s_wait_asynccnt 0 before exit.
// ---------------------------------------------------------------------------
__global__ void __launch_bounds__(256)
pp_zero_async_kernel(float* __restrict__ out, size_t n_chunks16)
{
    __shared__ __align__(16) float zbuf[256 * 4];   // 4KB of zeros per block

    const int tid = threadIdx.x;
    // Zero this lane's 16B staging slot, then make it visible block-wide.
    zbuf[tid * 4 + 0] = 0.0f;
    zbuf[tid * 4 + 1] = 0.0f;
    zbuf[tid * 4 + 2] = 0.0f;
    zbuf[tid * 4 + 3] = 0.0f;
    __syncthreads();   // emits s_wait_dscnt 0 + barrier: LDS zeros are committed

    // LDS byte offset of this lane's slot (AS3 ptrtoint == LDS offset).
    const uint32_t lds_off = (uint32_t)(uintptr_t)
        (__attribute__((address_space(3))) void*)(&zbuf[tid * 4]);

    const size_t stride = (size_t)gridDim.x * (size_t)blockDim.x;
    size_t i = (size_t)blockIdx.x * (size_t)blockDim.x + (size_t)tid;
    const uint64_t base = (uint64_t)(uintptr_t)out;

    for (; i < n_chunks16; i += stride) {
        const uint64_t gaddr = base + (i << 4);      // 16B per chunk
        asm volatile("global_store_async_from_lds_b128 %0, %1, off"
                     :: "v"(gaddr), "v"(lds_off)
                     : "memory");
    }

    // Drain ASYNCcnt so LDS can be safely deallocated / kernel retires clean.
    asm volatile("s_wait_asynccnt 0" ::: "memory");
}

// ---------------------------------------------------------------------------
// Kernel 2: scatter.  16 lanes per pillar; lane q handles channels 4q..4q+3.
// Feature loads are contiguous global_load_b128 (perfectly coalesced: 16
// lanes x 16B = one 256B pillar row).  Coord int4 loads broadcast within the
// wave and hit L0.  The 4B output stores are an inherent random scatter
// (stride G between channels, random g per pillar) -> global_store_b32.
// ---------------------------------------------------------------------------
__global__ void __launch_bounds__(256)
pp_scatter_kernel(const float* __restrict__ pf,
                  const int*   __restrict__ coords,
                  float*       __restrict__ out,
                  int n_pillars)
{
    const int t = blockIdx.x * blockDim.x + threadIdx.x;
    const int p = t >> 4;          // pillar row
    const int q = t & 15;          // channel group of 4
    if (p >= n_pillars) return;

    // coords row: [b, z, y, x]; reference idx = z + y*NX + x
    const int4   c4 = ((const int4*)coords)[p];
    const float4 f  = ((const float4*)pf)[(size_t)p * 16 + q];

    const int b = c4.x;
    const int g = c4.y + c4.z * PP_NX + c4.w;   // in [0, G)

    const int    cbase = q << 2;                // first of 4 channels
    size_t base = ((size_t)(b * PP_C + cbase)) * (size_t)PP_G + (size_t)g;

    out[base]                     = f.x;
    out[base + (size_t)PP_G]      = f.y;
    out[base + (size_t)2 * PP_G]  = f.z;
    out[base + (size_t)3 * PP_G]  = f.w;
}

// ---------------------------------------------------------------------------
// Host-side launcher (graph-capture safe: launches only, all on `stream`).
// ---------------------------------------------------------------------------
extern "C" void kernel_launch(void* const* d_in, const int* in_sizes, int n_in,
                              void* d_out, int out_size, void* d_ws, size_t ws_size,
                              hipStream_t stream)
{
    (void)n_in; (void)d_ws; (void)ws_size;

    const float* pf     = (const float*)d_in[0];   // (B*NPB, 64) f32
    const int*   coords = (const int*)d_in[1];     // (B*NPB, 4) i32
    float*       out    = (float*)d_out;           // (8, 64, 512, 512) f32

    // Pass 1: stream 512MB of zeros via async LDS->global b128 stores.
    const size_t n_chunks16 = (size_t)out_size / 4;   // float4 chunks
    pp_zero_async_kernel<<<8192, 256, 0, stream>>>(out, n_chunks16);

    // Pass 2: scatter pillars (16 lanes per pillar).
    const int n_pillars = in_sizes[1] / 4;            // 128000
    const int n_threads = n_pillars * 16;
    const int n_blocks  = (n_threads + 255) / 256;    // 8000
    pp_scatter_kernel<<<n_blocks, 256, 0, stream>>>(pf, coords, out, n_pillars);
}